// newLocalGlobalInfoNCE_23381801959614
// MI455X (gfx1250) — compile-verified
//
#include <hip/hip_runtime.h>
#include <math.h>

#define N_ROWS   25088
#define DIM      512
#define CLS      27
#define CPAD     32
#define SIMW     3136
#define INV_TEMP (1.0f / 0.07f)

typedef float v2f __attribute__((ext_vector_type(2)));
typedef float v8f __attribute__((ext_vector_type(8)));

// ---------------------------------------------------------------- zero scratch
__global__ void zero_kernel(float* p, int n) {
    int i = blockIdx.x * blockDim.x + threadIdx.x;
    if (i < n) p[i] = 0.0f;
}

// ---------------------------------------------------- weights = row-mean(sim)
// one wave32 per 3136-wide row; float4 loads (784 per row), xor-shuffle reduce
__global__ void weights_kernel(const float* __restrict__ sim, float* __restrict__ weights) {
    int wave = threadIdx.x >> 5;
    int lane = threadIdx.x & 31;
    int row  = blockIdx.x * 8 + wave;                     // grid = 3136, 8 waves/block
    const float4* p = (const float4*)(sim + (size_t)row * SIMW);
    float s = 0.0f;
    for (int k = lane; k < SIMW / 4; k += 32) {
        float4 v = p[k];
        s += (v.x + v.y) + (v.z + v.w);
    }
    for (int m = 16; m; m >>= 1) s += __shfl_xor(s, m, 32);
    if (lane == 0) weights[row] = s * (1.0f / (float)SIMW);
}

// ------------------------------------------------- segment sums via LDS stage
__global__ void seg_sum_kernel(const float* __restrict__ S1, const int* __restrict__ seg,
                               float* __restrict__ sums, float* __restrict__ counts) {
    __shared__ float lsum[CLS * DIM];   // 55296 B
    __shared__ float lcnt[CLS];
    int tid = threadIdx.x;              // 256 threads
    for (int i = tid; i < CLS * DIM; i += 256) lsum[i] = 0.0f;
    if (tid < CLS) lcnt[tid] = 0.0f;
    __syncthreads();

    int row0 = blockIdx.x * 256;        // grid = 98 -> 25088 rows
    for (int r = 0; r < 256; ++r) {
        int row = row0 + r;
        int c = seg[row];
        float2 v = ((const float2*)(S1 + (size_t)row * DIM))[tid];
        atomicAdd(&lsum[c * DIM + 2 * tid],     v.x);   // ds_add_f32
        atomicAdd(&lsum[c * DIM + 2 * tid + 1], v.y);
        if (tid == 0) lcnt[c] += 1.0f;
    }
    __syncthreads();
    for (int i = tid; i < CLS * DIM; i += 256) atomicAdd(&sums[i], lsum[i]);
    if (tid < CLS) atomicAdd(&counts[tid], lcnt[tid]);
}

// ----------------------- centroids = sums / max(count,1) * (1/TEMP), in place
__global__ void centroid_kernel(float* __restrict__ sums, const float* __restrict__ counts) {
    int i = blockIdx.x * 256 + threadIdx.x;               // 32*512 = 16384
    int c = i >> 9;
    float cnt = (c < CLS) ? counts[c] : 0.0f;
    sums[i] = sums[i] * (INV_TEMP / fmaxf(cnt, 1.0f));    // pad rows stay 0
}

// --------------------------- fused WMMA dual-GEMM + cross-entropy + reduction
// one wave per 16-row chunk; fp32 WMMA 16x16x4; two 16-wide class tiles (27->32)
__global__ void gemm_ce_kernel(const float* __restrict__ S1, const float* __restrict__ S2,
                               const float* __restrict__ cent, const int* __restrict__ seg,
                               const float* __restrict__ weights, float* __restrict__ lossAcc) {
    int wave = threadIdx.x >> 5;
    int lane = threadIdx.x & 31;
    int half = lane >> 4;          // K-pair select for A/B fragments
    int n16  = lane & 15;
    int chunk = blockIdx.x * 8 + wave;                    // grid = 196 -> 1568 chunks
    int m0 = chunk * 16;

    // fp32 A layout (16x4): lanes 0-15 hold (K=0,1) of row=lane; lanes 16-31 hold (K=2,3)
    // of row=lane-16  => contiguous float2 per lane.  Symmetric B layout assumed:
    // B[k][n] at lane = n + 16*(k>=2), vgpr = k&1 => contiguous float2 of centroids row n.
    int kOff = half * 2;
    const v2f* a1p = (const v2f*)(S1   + (size_t)(m0 + n16) * DIM + kOff);
    const v2f* a2p = (const v2f*)(S2   + (size_t)(m0 + n16) * DIM + kOff);
    const v2f* b0p = (const v2f*)(cent + (size_t)(n16)      * DIM + kOff);
    const v2f* b1p = (const v2f*)(cent + (size_t)(n16 + 16) * DIM + kOff);

    v8f al0 = {}, al1 = {}, ag0 = {}, ag1 = {};
    for (int k = 0; k < DIM; k += 4) {
        v2f a1 = a1p[k >> 1];
        v2f a2 = a2p[k >> 1];
        v2f b0 = b0p[k >> 1];
        v2f b1 = b1p[k >> 1];
        al0 = __builtin_amdgcn_wmma_f32_16x16x4_f32(false, a1, false, b0, (short)0, al0, false, false);
        al1 = __builtin_amdgcn_wmma_f32_16x16x4_f32(false, a1, false, b1, (short)0, al1, false, false);
        ag0 = __builtin_amdgcn_wmma_f32_16x16x4_f32(false, a2, false, b0, (short)0, ag0, false, false);
        ag1 = __builtin_amdgcn_wmma_f32_16x16x4_f32(false, a2, false, b1, (short)0, ag1, false, false);
    }

    // C/D layout: vgpr v, lane L -> row M = v + 8*(L>=16), class N = (L&15) + 16*tile.
    // CE over 27 classes: tile1 lanes with class >= 27 masked to -inf.
    const float NEG_INF = -__builtin_inff();
    bool valid1 = (n16 < (CLS - 16));
    float pl = 0.0f, pg = 0.0f;
    int rowBase = m0 + half * 8;
#pragma unroll
    for (int v = 0; v < 8; ++v) {
        int row = rowBase + v;
        int label = seg[row];
        float zl0 = al0[v], zl1 = valid1 ? al1[v] : NEG_INF;
        float zg0 = ag0[v], zg1 = valid1 ? ag1[v] : NEG_INF;

        float ml = fmaxf(zl0, zl1), mg = fmaxf(zg0, zg1);
        for (int m = 1; m < 16; m <<= 1) {
            ml = fmaxf(ml, __shfl_xor(ml, m, 32));
            mg = fmaxf(mg, __shfl_xor(mg, m, 32));
        }
        float el = expf(zl0 - ml) + expf(zl1 - ml);   // exp(-inf)=0 handles padding
        float eg = expf(zg0 - mg) + expf(zg1 - mg);
        for (int m = 1; m < 16; m <<= 1) {
            el += __shfl_xor(el, m, 32);
            eg += __shfl_xor(eg, m, 32);
        }
        float sl = (n16 == label) ? zl0 : 0.0f; if (n16 + 16 == label) sl = zl1;
        float sg = (n16 == label) ? zg0 : 0.0f; if (n16 + 16 == label) sg = zg1;
        for (int m = 1; m < 16; m <<= 1) {
            sl += __shfl_xor(sl, m, 32);
            sg += __shfl_xor(sg, m, 32);
        }
        float w = weights[row];
        pl += w * (logf(el) + ml - sl);
        pg += w * (logf(eg) + mg - sg);
    }
    if (n16 == 0) {                        // lanes 0 & 16: one partial per 8 rows
        atomicAdd(&lossAcc[0], pl);
        atomicAdd(&lossAcc[1], pg);
    }
}

// -------------------------------------------------------------------- finalize
__global__ void finalize_kernel(const float* __restrict__ lossAcc, float* __restrict__ out) {
    // total = (0.5*local_mean + 0.5*global_mean)/2 = (sum_l + sum_g) / (4N)
    out[0] = (lossAcc[0] + lossAcc[1]) * (1.0f / (4.0f * (float)N_ROWS));
}

extern "C" void kernel_launch(void* const* d_in, const int* in_sizes, int n_in,
                              void* d_out, int out_size, void* d_ws, size_t ws_size,
                              hipStream_t stream) {
    const float* S1  = (const float*)d_in[0];
    const float* S2  = (const float*)d_in[1];
    const int*   seg = (const int*)d_in[2];
    const float* sim = (const float*)d_in[3];
    float* out = (float*)d_out;

    float* ws      = (float*)d_ws;
    float* weights = ws;                       // 25088
    float* cent    = ws + N_ROWS;              // 32*512 (sums -> centroids in place)
    float* counts  = cent + CPAD * DIM;        // 32
    float* lossAcc = counts + CPAD;            // 2

    int nzero = CPAD * DIM + CPAD + 2;         // cent + counts + lossAcc, contiguous
    zero_kernel    <<<(nzero + 255) / 256, 256, 0, stream>>>(cent, nzero);
    weights_kernel <<<N_ROWS / 8, 256, 0, stream>>>(sim, weights);
    seg_sum_kernel <<<N_ROWS / 256, 256, 0, stream>>>(S1, seg, cent, counts);
    centroid_kernel<<<(CPAD * DIM) / 256, 256, 0, stream>>>(cent, counts);
    gemm_ce_kernel <<<N_ROWS / 128, 256, 0, stream>>>(S1, S2, cent, seg, weights, lossAcc);
    finalize_kernel<<<1, 1, 0, stream>>>(lossAcc, out);
}